// cuEquivInteraction_56495999812197
// MI455X (gfx1250) — compile-verified
//
#include <hip/hip_runtime.h>
#include <math.h>

// ---------------------------------------------------------------------------
// MI455X (gfx1250) implementation of the equivariant interaction block.
// Memory/atomic-bound workload -> keep f32 everywhere, use f32 WMMA
// (V_WMMA_F32_16X16X4_F32) for the GEMM-shaped stages, fuse the radial MLP
// into the edge kernel (never materialize Rl = 282MB), scatter with
// GLOBAL_ATOMIC_ADD_F32 into an L2-resident 23MB accumulator.
// Out-of-range GEMM rows are CLAMPED (not predicated): a WMMA A-row only
// feeds the same D-row, so garbage rows are harmless as long as stores are
// guarded -- this avoids per-element exec-masked loads.
// ---------------------------------------------------------------------------

typedef float v2f __attribute__((ext_vector_type(2)));
typedef float v8f __attribute__((ext_vector_type(8)));

#define NF        576      // 9 * 64 node feature width
#define NOUTC     256      // 4 * 64 output width
#define RLW       704      // 11 paths * 64 channels
#define EPB       16       // edges per block (edge kernel)
#define INV_SQRT_K 0.125f
#define MIX_NORM   0.03952847075210474f   // 1/sqrt(64*10)
#define INV_AVG    0.1f                   // 1/AVG_NEIGH

__device__ inline v8f wmma4(v2f a, v2f b, v8f c) {
  // D = A(16x4,f32) * B(4x16,f32) + C(16x16,f32)
  return __builtin_amdgcn_wmma_f32_16x16x4_f32(false, a, false, b, (short)0, c,
                                               false, false);
}

__device__ inline float silu(float x) { return x / (1.f + __expf(-x)); }

// ---------------- real-SH Clebsch-Gordan sparse tables ---------------------
// l=1 real basis order (y,z,x); l=2 order (xy,yz,z2,zx,x2-y2).
// Values from offline real-CG tabulation (orthonormal CG normalization).
struct CGE { int a, b, c; float v; };
struct EP  { int aoff, boff, ooff, nm, off, cnt; };

static constexpr float S13 = 0.57735027f;   // 1/sqrt(3)
static constexpr float S15 = 0.44721360f;   // 1/sqrt(5)
static constexpr float S12 = 0.70710678f;   // 1/sqrt(2)
static constexpr float S16 = 0.40824829f;   // 1/sqrt(6)
static constexpr float T12 = 0.54772256f;   // sqrt(3/10)
static constexpr float T16 = 0.31622777f;
static constexpr float T26 = 0.63245553f;
static constexpr float U1  = 0.26726124f;   // 1/sqrt(14)
static constexpr float U2  = 0.53452248f;   // 2/sqrt(14)
static constexpr float U3  = 0.46291005f;   // sqrt(3/14)

static constexpr CGE CG_TAB[] = {
  // [0]   P000 (1)
  {0,0,0,1.f},
  // [1]   P011 (3)
  {0,0,0,1.f},{0,1,1,1.f},{0,2,2,1.f},
  // [4]   P022 (5)
  {0,0,0,1.f},{0,1,1,1.f},{0,2,2,1.f},{0,3,3,1.f},{0,4,4,1.f},
  // [9]   P101 (3)
  {0,0,0,1.f},{1,0,1,1.f},{2,0,2,1.f},
  // [12]  P110 (3)
  {0,0,0,S13},{1,1,0,S13},{2,2,0,S13},
  // [15]  P112 (11)
  {2,0,0,S12},{0,2,0,S12},
  {0,1,1,S12},{1,0,1,S12},
  {0,0,2,-S16},{2,2,2,-S16},{1,1,2,2.f*S16},
  {1,2,3,S12},{2,1,3,S12},
  {2,2,4,S12},{0,0,4,-S12},
  // [26]  P121 (11)
  {2,0,0,T12},{0,0,2,T12},
  {0,1,1,T12},{1,1,0,T12},
  {0,2,0,-T16},{2,2,2,-T16},{1,2,1,T26},
  {1,3,2,T12},{2,3,1,T12},
  {2,4,2,T12},{0,4,0,-T12},
  // [37]  P202 (5)
  {0,0,0,1.f},{1,0,1,1.f},{2,0,2,1.f},{3,0,3,1.f},{4,0,4,1.f},
  // [42]  P211 (11)
  {0,2,0,T12},{0,0,2,T12},
  {1,0,1,T12},{1,1,0,T12},
  {2,0,0,-T16},{2,2,2,-T16},{2,1,1,T26},
  {3,1,2,T12},{3,2,1,T12},
  {4,2,2,T12},{4,0,0,-T12},
  // [53]  P220 (5)
  {0,0,0,S15},{1,1,0,S15},{2,2,0,S15},{3,3,0,S15},{4,4,0,S15},
  // [58]  P222 (20)
  {0,0,2,-U2},{1,1,2,U1},{3,3,2,U1},{4,4,2,-U2},{2,2,2,U2},
  {0,2,0,U1},{2,0,0,U1},{1,3,0,U3},{3,1,0,U3},
  {1,2,1,U1},{2,1,1,U1},{0,3,1,U3},{3,0,1,U3},
  {3,2,3,U1},{2,3,3,U1},{0,1,3,-U3},{1,0,3,-U3},
  {4,2,4,U1},{2,4,4,U1},{1,1,4,U3},
};

// edge paths in the reference's enumeration order
// aoff: sph m-offset {l:0->0,1->1,2->4}; boff/ooff: 9-row offsets {0,1,4}
static constexpr EP EPATHS[11] = {
  {0,0,0,1, 0,1},   // (0,0,0)
  {0,1,1,3, 1,3},   // (0,1,1)
  {0,4,4,5, 4,5},   // (0,2,2)
  {1,0,1,3, 9,3},   // (1,0,1)
  {1,1,0,1,12,3},   // (1,1,0)
  {1,1,4,5,15,11},  // (1,1,2)
  {1,4,1,3,26,11},  // (1,2,1)
  {4,0,4,5,37,5},   // (2,0,2)
  {4,1,1,3,42,11},  // (2,1,1)
  {4,4,0,1,53,5},   // (2,2,0)
  {4,4,4,5,58,20},  // (2,2,2)
};
// degree-2 / degree-3 contraction paths; a/b offsets in row spaces, out in
// 4-row (L=0 row0, L=1 rows1..3) space.
static constexpr EP D2P[5] = {
  {0,0,0,1, 0,1}, {1,1,0,1,12,3}, {4,4,0,1,53,5},
  {0,1,1,3, 1,3}, {1,4,1,3,26,11},
};
static constexpr EP D3P[5] = {
  {0,0,0,1, 0,1}, {1,1,0,1,12,3}, {0,1,1,3, 1,3},
  {1,0,1,3, 9,3}, {1,4,1,3,26,11},
};

// ---------------------------------------------------------------------------
// K1: h1p[n,576] = per-l (h1 @ W_lin_in[l]) * INV_SQRT_K   (WMMA f32 16x16x4)
// rows domain per l: l0 -> N rows, l1 -> 3N, l2 -> 5N
// ---------------------------------------------------------------------------
__global__ void k_lin_h1p(const float* __restrict__ x,
                          const float* __restrict__ W,
                          float* __restrict__ y, int N) {
  int wave = blockIdx.x * (blockDim.x >> 5) + (threadIdx.x >> 5);
  int lane = threadIdx.x & 31;
  int t0 = (N + 15) / 16, t1 = (3 * N + 15) / 16, t2 = (5 * N + 15) / 16;
  int l, r0;
  if (wave < t0)              { l = 0; r0 = wave * 16; }
  else if (wave < t0 + t1)    { l = 1; r0 = (wave - t0) * 16; }
  else if (wave < t0 + t1 + t2) { l = 2; r0 = (wave - t0 - t1) * 16; }
  else return;
  int mul = 2 * l + 1;
  int seg = (l == 0) ? 0 : ((l == 1) ? 64 : 256);
  int rows = N * mul;
  int hi = lane >> 4, rr = lane & 15;
  // clamp out-of-range rows: they only feed unstored D-rows
  int r = r0 + rr; if (r >= rows) r = rows - 1;
  int n = r / mul, m = r % mul;
  const float* arow = x + (size_t)n * NF + seg + m * 64;
  const float* Wl = W + (size_t)l * 64 * 64;

  v8f acc[4] = {};
#pragma unroll
  for (int k0 = 0; k0 < 64; k0 += 4) {
    int ka = k0 + 2 * hi;
    v2f a; a[0] = arow[ka]; a[1] = arow[ka + 1];
#pragma unroll
    for (int ct = 0; ct < 4; ct++) {
      int col = ct * 16 + rr;
      v2f b; b[0] = Wl[ka * 64 + col]; b[1] = Wl[(ka + 1) * 64 + col];
      acc[ct] = wmma4(a, b, acc[ct]);
    }
  }
#pragma unroll
  for (int ct = 0; ct < 4; ct++)
#pragma unroll
    for (int i = 0; i < 8; i++) {
      int row = r0 + i + 8 * hi;
      if (row < rows) {
        int nn = row / mul, mm = row % mul;
        y[(size_t)nn * NF + seg + mm * 64 + ct * 16 + rr] =
            acc[ct][i] * INV_SQRT_K;
      }
    }
}

// ---------------------------------------------------------------------------
// K2: fused edge stage. Per block (1 wave, 16 edges):
//   A.1 per-lane residual radial MLP (scalar weights)      -> H[16][32] in LDS
//   A.2 Rl tile [16,704] = H @ mlp_W3 + b3 via WMMA f32    -> RL in LDS
//   B   CG tensor product per edge (constexpr-unrolled) + atomic scatter to A
// ---------------------------------------------------------------------------
__global__ void __launch_bounds__(32)
k_edge(const float* __restrict__ elen, const float* __restrict__ esph,
       const int* __restrict__ sender, const int* __restrict__ receiver,
       const float* __restrict__ W0, const float* __restrict__ b0,
       const float* __restrict__ W1, const float* __restrict__ b1,
       const float* __restrict__ W2, const float* __restrict__ b2,
       const float* __restrict__ W3, const float* __restrict__ b3,
       const float* __restrict__ h1p, float* __restrict__ A, int E) {
  __shared__ float H[EPB][33];        // padded: conflict-free WMMA A reads
  __shared__ float RL[EPB][RLW + 4];
  int lane = threadIdx.x;
  int e0 = blockIdx.x * EPB;

  // ---- A.1: residual MLP, one edge per lane (lanes 0..15)
  if (lane < EPB) {
    int e = e0 + lane; if (e >= E) e = E - 1;
    float xin[8];
#pragma unroll
    for (int i = 0; i < 8; i++) xin[i] = elen[(size_t)e * 8 + i];
    float h[32];
#pragma unroll
    for (int c = 0; c < 32; c++) {
      float s = b0[c];
#pragma unroll
      for (int i = 0; i < 8; i++) s += xin[i] * W0[i * 32 + c];
      h[c] = silu(s);
    }
    float g1[32];
#pragma unroll
    for (int c = 0; c < 32; c++) {
      float s = b1[c];
#pragma unroll
      for (int i = 0; i < 32; i++) s += h[i] * W1[i * 32 + c];
      g1[c] = h[c] + silu(s);
    }
    float g2[32];
#pragma unroll
    for (int c = 0; c < 32; c++) {
      float s = b2[c];
#pragma unroll
      for (int i = 0; i < 32; i++) s += g1[i] * W2[i * 32 + c];
      g2[c] = g1[c] + silu(s);
    }
#pragma unroll
    for (int c = 0; c < 32; c++) H[lane][c] = g2[c];
  }
  __syncthreads();

  // ---- A.2: Rl = H(16x32) @ W3(32x704) + b3 via WMMA f32 16x16x4
  int hi = lane >> 4, rr = lane & 15;
#pragma unroll 2
  for (int ct = 0; ct < RLW / 16; ct++) {
    int col = ct * 16 + rr;
    v8f acc = {};
#pragma unroll
    for (int k0 = 0; k0 < 32; k0 += 4) {
      int ka = k0 + 2 * hi;
      v2f a; a[0] = H[rr][ka];              a[1] = H[rr][ka + 1];
      v2f b; b[0] = W3[(size_t)ka * RLW + col];
             b[1] = W3[(size_t)(ka + 1) * RLW + col];
      acc = wmma4(a, b, acc);
    }
    float bias = b3[col];
#pragma unroll
    for (int i = 0; i < 8; i++) RL[i + 8 * hi][col] = acc[i] + bias;
  }
  __syncthreads();

  // ---- B: channelwise CG tensor product + atomic scatter
  // lane handles channels (lane, lane+32)
  for (int ei = 0; ei < EPB; ei++) {
    int e = e0 + ei;
    if (e >= E) break;
    int snd = sender[e], rcv = receiver[e];
    float sph[9];
#pragma unroll
    for (int i = 0; i < 9; i++) sph[i] = esph[(size_t)e * 9 + i];
    float g[9][2];
    const float* hp = h1p + (size_t)snd * NF;
#pragma unroll
    for (int row = 0; row < 9; row++) {
      g[row][0] = hp[row * 64 + lane];
      g[row][1] = hp[row * 64 + lane + 32];
    }
    float msg[9][2];
#pragma unroll
    for (int row = 0; row < 9; row++) { msg[row][0] = 0.f; msg[row][1] = 0.f; }
#pragma unroll
    for (int p = 0; p < 11; p++) {
      const EP pi = EPATHS[p];
      float t[5][2] = {};
#pragma unroll
      for (int q = 0; q < pi.cnt; q++) {
        const CGE en = CG_TAB[pi.off + q];
        float sv = en.v * sph[pi.aoff + en.a];
        t[en.c][0] += sv * g[pi.boff + en.b][0];
        t[en.c][1] += sv * g[pi.boff + en.b][1];
      }
      float rl0 = RL[ei][p * 64 + lane];
      float rl1 = RL[ei][p * 64 + lane + 32];
#pragma unroll
      for (int d = 0; d < pi.nm; d++) {
        msg[pi.ooff + d][0] += t[d][0] * rl0;
        msg[pi.ooff + d][1] += t[d][1] * rl1;
      }
    }
    float* Ab = A + (size_t)rcv * NF;
#pragma unroll
    for (int row = 0; row < 9; row++) {
      atomicAdd(&Ab[row * 64 + lane],      msg[row][0]);
      atomicAdd(&Ab[row * 64 + lane + 32], msg[row][1]);
    }
  }
}

// ---------------------------------------------------------------------------
// K3: per-node element-conditioned mixing + symmetric contraction -> B[N,256]
// one wave per node; lane handles channels (lane, lane+32)
// ---------------------------------------------------------------------------
__global__ void k_node(const float* __restrict__ Araw,
                       const int* __restrict__ indices,
                       const float* __restrict__ Wmix,
                       const float* __restrict__ w1,
                       const float* __restrict__ w2,
                       const float* __restrict__ w3,
                       float* __restrict__ Bout, int N) {
  int wv = threadIdx.x >> 5, lane = threadIdx.x & 31;
  int n = blockIdx.x * (blockDim.x >> 5) + wv;
  if (n >= N) return;
  int e = indices[n];

  float a[9][2];
  const float* Ab = Araw + (size_t)n * NF;
#pragma unroll
  for (int row = 0; row < 9; row++) {
    a[row][0] = Ab[row * 64 + lane] * INV_AVG;
    a[row][1] = Ab[row * 64 + lane + 32] * INV_AVG;
  }
  // Amix[l] = A[l] @ W_mix[l][e] * MIX_NORM   (shfl-broadcast GEMV)
  float am[9][2];
  const int roff[3] = {0, 1, 4};
#pragma unroll
  for (int l = 0; l < 3; l++) {
    const float* Wl = Wmix + (((size_t)l * 10 + e) * 64) * 64;
#pragma unroll
    for (int mi = 0; mi < 2 * l + 1; mi++) {
      int row = roff[l] + mi;
      float acc0 = 0.f, acc1 = 0.f;
#pragma unroll
      for (int half = 0; half < 2; half++) {
        float src = a[row][half];
        for (int k = 0; k < 32; k++) {
          float av = __shfl(src, k, 32);
          const float* wr = Wl + (half * 32 + k) * 64;
          acc0 += av * wr[lane];
          acc1 += av * wr[lane + 32];
        }
      }
      am[row][0] = acc0 * MIX_NORM;
      am[row][1] = acc1 * MIX_NORM;
    }
  }
  // T2 (4 out rows: L0 row0, L1 rows1..3)
  float t2[4][2] = {};
#pragma unroll
  for (int p = 0; p < 5; p++) {
    const EP pi = D2P[p];
    float tt[3][2] = {};
#pragma unroll
    for (int q = 0; q < pi.cnt; q++) {
      const CGE en = CG_TAB[pi.off + q];
      tt[en.c][0] += en.v * am[pi.aoff + en.a][0] * am[pi.boff + en.b][0];
      tt[en.c][1] += en.v * am[pi.aoff + en.a][1] * am[pi.boff + en.b][1];
    }
    float w20 = w2[((size_t)e * 5 + p) * 64 + lane];
    float w21 = w2[((size_t)e * 5 + p) * 64 + lane + 32];
#pragma unroll
    for (int d = 0; d < pi.nm; d++) {
      t2[pi.ooff + d][0] += w20 * tt[d][0];
      t2[pi.ooff + d][1] += w21 * tt[d][1];
    }
  }
  // T3
  float t3[4][2] = {};
#pragma unroll
  for (int q3 = 0; q3 < 5; q3++) {
    const EP pi = D3P[q3];
    float tt[3][2] = {};
#pragma unroll
    for (int q = 0; q < pi.cnt; q++) {
      const CGE en = CG_TAB[pi.off + q];
      tt[en.c][0] += en.v * t2[pi.aoff + en.a][0] * am[pi.boff + en.b][0];
      tt[en.c][1] += en.v * t2[pi.aoff + en.a][1] * am[pi.boff + en.b][1];
    }
    float w30 = w3[((size_t)e * 5 + q3) * 64 + lane];
    float w31 = w3[((size_t)e * 5 + q3) * 64 + lane + 32];
#pragma unroll
    for (int d = 0; d < pi.nm; d++) {
      t3[pi.ooff + d][0] += w30 * tt[d][0];
      t3[pi.ooff + d][1] += w31 * tt[d][1];
    }
  }
  // B = w1 * Amix + T2 + T3  (rows: L0 row0, L1 rows1..3)
  float* Bo = Bout + (size_t)n * NOUTC;
#pragma unroll
  for (int L = 0; L < 2; L++) {
    float wa = w1[((size_t)e * 2 + L) * 64 + lane];
    float wb = w1[((size_t)e * 2 + L) * 64 + lane + 32];
    int nm = (L == 0) ? 1 : 3;
#pragma unroll
    for (int d = 0; d < 3; d++) {
      if (d >= nm) break;
      int row = L + d;           // roL = {0,1}
      float b0v = wa * am[row][0] + t2[row][0] + t3[row][0];
      float b1v = wb * am[row][1] + t2[row][1] + t3[row][1];
      Bo[row * 64 + lane]      = b0v;
      Bo[row * 64 + lane + 32] = b1v;
    }
  }
}

// ---------------------------------------------------------------------------
// K4: out = (h1 @ W_out_h[L] + B @ W_out_m[L]) * INV_SQRT_K  (WMMA f32)
// rows: L=0 -> N rows, L=1 -> 3N rows; out layout [N,256]
// ---------------------------------------------------------------------------
__global__ void k_out(const float* __restrict__ x, const float* __restrict__ B,
                      const float* __restrict__ Woh,
                      const float* __restrict__ Wom,
                      float* __restrict__ y, int N) {
  int wave = blockIdx.x * (blockDim.x >> 5) + (threadIdx.x >> 5);
  int lane = threadIdx.x & 31;
  int t0 = (N + 15) / 16, t1 = (3 * N + 15) / 16;
  int L, r0;
  if (wave < t0)           { L = 0; r0 = wave * 16; }
  else if (wave < t0 + t1) { L = 1; r0 = (wave - t0) * 16; }
  else return;
  int mul = (L == 0) ? 1 : 3;
  int seg = (L == 0) ? 0 : 64;
  int rows = N * mul;
  int hi = lane >> 4, rr = lane & 15;
  // clamp out-of-range rows (see header comment)
  int r = r0 + rr; if (r >= rows) r = rows - 1;
  int n = r / mul, m = r % mul;
  const float* ah = x + (size_t)n * NF + seg + m * 64;
  const float* ab = B + (size_t)n * NOUTC + seg + m * 64;
  const float* Wh = Woh + (size_t)L * 64 * 64;
  const float* Wm = Wom + (size_t)L * 64 * 64;

  v8f acc[4] = {};
#pragma unroll
  for (int k0 = 0; k0 < 64; k0 += 4) {
    int ka = k0 + 2 * hi;
    v2f a1; a1[0] = ah[ka]; a1[1] = ah[ka + 1];
    v2f a2; a2[0] = ab[ka]; a2[1] = ab[ka + 1];
#pragma unroll
    for (int ct = 0; ct < 4; ct++) {
      int col = ct * 16 + rr;
      v2f b1; b1[0] = Wh[ka * 64 + col]; b1[1] = Wh[(ka + 1) * 64 + col];
      acc[ct] = wmma4(a1, b1, acc[ct]);
      v2f b2; b2[0] = Wm[ka * 64 + col]; b2[1] = Wm[(ka + 1) * 64 + col];
      acc[ct] = wmma4(a2, b2, acc[ct]);
    }
  }
#pragma unroll
  for (int ct = 0; ct < 4; ct++)
#pragma unroll
    for (int i = 0; i < 8; i++) {
      int row = r0 + i + 8 * hi;
      if (row < rows) {
        int nn = row / mul, mm = row % mul;
        y[(size_t)nn * NOUTC + seg + mm * 64 + ct * 16 + rr] =
            acc[ct][i] * INV_SQRT_K;
      }
    }
}

// ---------------------------------------------------------------------------
extern "C" void kernel_launch(void* const* d_in, const int* in_sizes, int n_in,
                              void* d_out, int out_size, void* d_ws,
                              size_t ws_size, hipStream_t stream) {
  const int*   sender  = (const int*)d_in[0];
  const int*   receiver= (const int*)d_in[1];
  const int*   indices = (const int*)d_in[2];
  // d_in[3] = node_attrs (one-hot) -- replaced by `indices`
  const float* nfi     = (const float*)d_in[4];
  const float* elen    = (const float*)d_in[5];
  const float* esph    = (const float*)d_in[6];
  const float* Wlin    = (const float*)d_in[7];
  const float* W0 = (const float*)d_in[8];  const float* b0 = (const float*)d_in[9];
  const float* W1 = (const float*)d_in[10]; const float* b1 = (const float*)d_in[11];
  const float* W2 = (const float*)d_in[12]; const float* b2 = (const float*)d_in[13];
  const float* W3 = (const float*)d_in[14]; const float* b3 = (const float*)d_in[15];
  const float* Wmix = (const float*)d_in[16];
  const float* w1   = (const float*)d_in[17];
  const float* w2   = (const float*)d_in[18];
  const float* w3   = (const float*)d_in[19];
  const float* Wom  = (const float*)d_in[20];
  const float* Woh  = (const float*)d_in[21];
  (void)n_in; (void)out_size; (void)ws_size;

  int E = in_sizes[0];
  int N = in_sizes[4] / NF;

  float* h1p  = (float*)d_ws;
  float* Aacc = h1p + (size_t)N * NF;
  float* Bws  = Aacc + (size_t)N * NF;

  (void)hipMemsetAsync(Aacc, 0, (size_t)N * NF * sizeof(float), stream);

  int wv1 = (N + 15) / 16 + (3 * N + 15) / 16 + (5 * N + 15) / 16;
  k_lin_h1p<<<(wv1 + 3) / 4, 128, 0, stream>>>(nfi, Wlin, h1p, N);

  k_edge<<<(E + EPB - 1) / EPB, 32, 0, stream>>>(
      elen, esph, sender, receiver, W0, b0, W1, b1, W2, b2, W3, b3,
      h1p, Aacc, E);

  k_node<<<(N + 7) / 8, 256, 0, stream>>>(Aacc, indices, Wmix, w1, w2, w3,
                                          Bws, N);

  int wv4 = (N + 15) / 16 + (3 * N + 15) / 16;
  k_out<<<(wv4 + 3) / 4, 128, 0, stream>>>(nfi, Bws, Woh, Wom, (float*)d_out,
                                           N);
}